// AQIGraphModel_566935683142
// MI455X (gfx1250) — compile-verified
//
#include <hip/hip_runtime.h>

// ---------------------------------------------------------------------------
// 3-layer GCN (PyG GCNConv semantics) for MI455X / gfx1250.
//   - degree + rsqrt computed ONCE, reused by all 3 layers
//   - node-feature GEMMs: V_WMMA_F32_16X16X4_F32, one wave per 16x16 D-tile.
//     A operand: one unconditional global_load_b64 per K-step (K padded to 4).
//     B operand: weights pre-swizzled into WMMA lane order -> one b64 load.
//   - edge scatter: float4 gathers + global_atomic_add_f32 (agg lives in L2)
//   - self-loop + bias + ReLU fused, in place
// Memory-bound: ~0.9 GB traffic -> ~40us at 23.3 TB/s HBM; GEMMs negligible.
// ---------------------------------------------------------------------------

typedef float v2f __attribute__((ext_vector_type(2)));
typedef float v8f __attribute__((ext_vector_type(8)));

// ---- utility ---------------------------------------------------------------
__global__ void k_fill(float* __restrict__ p, float v, int n) {
  int i = blockIdx.x * blockDim.x + threadIdx.x;
  if (i < n) p[i] = v;
}

// deg[dst] += 1 per edge (deg pre-filled with 1.0 for the self loop)
__global__ void k_deg(const long long* __restrict__ dst, float* __restrict__ deg,
                      long long E) {
  long long e = (long long)blockIdx.x * blockDim.x + threadIdx.x;
  if (e < E) atomicAdd(&deg[dst[e]], 1.0f);
}

// deg -> dinv in place (deg >= 1 always because of self loops)
__global__ void k_rsqrt(float* __restrict__ p, int n) {
  int i = blockIdx.x * blockDim.x + threadIdx.x;
  if (i < n) p[i] = rsqrtf(p[i]);
}

// pack x[N,3] -> xp[N,4], zero-padded K
__global__ void k_packX(const float* __restrict__ x, float* __restrict__ xp, int N) {
  int idx = blockIdx.x * blockDim.x + threadIdx.x;
  if (idx >= N * 4) return;
  int i = idx >> 2, c = idx & 3;
  xp[idx] = (c < 3) ? x[i * 3 + c] : 0.0f;
}

// Pre-swizzle W[K,C] into WMMA B-operand order, zero-padded in K and C:
//   Wp[((nt*KQ + kq)*32 + lane)*2 + j] = W[k*C + col]  (0 if OOB)
//   k   = kq*4 + (lane>>4)*2 + j      (f32 B layout: VGPR j holds K-rows
//   col = nt*16 + (lane&15)            {j, j+2} split across lane halves)
__global__ void k_packW(const float* __restrict__ W, float* __restrict__ Wp,
                        int K, int C, int KQ, int NT) {
  int idx = blockIdx.x * blockDim.x + threadIdx.x;
  int tot = NT * KQ * 32 * 2;
  if (idx >= tot) return;
  int j    = idx & 1;
  int lane = (idx >> 1) & 31;
  int kq   = (idx >> 6) % KQ;
  int nt   = (idx >> 6) / KQ;
  int k    = kq * 4 + ((lane >> 4) << 1) + j;
  int col  = nt * 16 + (lane & 15);
  Wp[idx] = (k < K && col < C) ? W[k * C + col] : 0.0f;
}

// ---- WMMA GEMM: H[N,Cout] = A[N,K] @ W[K,*]  (Wp pre-swizzled) -------------
// One wave per 16x16 tile; inner loop = 2x global_load_b64 + 1x wmma.
__global__ void k_gemm_wmma(const float* __restrict__ A, const float* __restrict__ Wp,
                            float* __restrict__ H, int N, int K, int Cout,
                            int ntilesN, int mtiles, int KQ) {
  int gwave = (int)(((long long)blockIdx.x * blockDim.x + threadIdx.x) >> 5);
  int lane  = threadIdx.x & 31;
  int mtile = gwave / ntilesN;
  int ntile = gwave - mtile * ntilesN;
  if (mtile >= mtiles) return;            // wave-uniform: EXEC all-1s at WMMA

  int r = mtile * 16 + (lane & 15);
  if (r >= N) r = N - 1;                  // clamp (branch-free); store is guarded
  const float* ap = A + (long long)r * K + ((lane >> 4) << 1);
  const float* bp = Wp + (((long long)ntile * KQ) << 6) + (lane << 1);

  v8f acc = {};
  for (int kq = 0; kq < KQ; ++kq) {
    v2f a = *(const v2f*)(ap + (kq << 2));   // global_load_b64
    v2f b = *(const v2f*)(bp + (kq << 6));   // global_load_b64
    acc = __builtin_amdgcn_wmma_f32_16x16x4_f32(
        false, a, false, b, (short)0, acc, false, false);
  }

  int col = ntile * 16 + (lane & 15);
  int rb  = mtile * 16 + ((lane >> 4) << 3);
  if (col < Cout) {
#pragma unroll
    for (int v = 0; v < 8; ++v)
      if (rb + v < N) H[(long long)(rb + v) * Cout + col] = acc[v];
  }
}

// ---- edge scatter: agg[dst] += h[src] * dinv[src]*dinv[dst] ----------------
__global__ void k_scatter(const long long* __restrict__ src,
                          const long long* __restrict__ dst,
                          const float* __restrict__ dinv,
                          const float* __restrict__ h,
                          float* __restrict__ agg, long long E, int C) {
  long long e = (long long)blockIdx.x * blockDim.x + threadIdx.x;
  if (e >= E) return;
  long long s = src[e], d = dst[e];
  const float* hs = h + s * C;
  __builtin_prefetch(hs, 0, 0);           // global_prefetch_b8 on gather row
  float norm = dinv[s] * dinv[d];
  float* ad = agg + d * C;
  for (int c = 0; c < C; c += 4) {        // C is 32/16/8 -> always /4
    float4 v = *(const float4*)(hs + c);
    atomicAdd(ad + c + 0, v.x * norm);
    atomicAdd(ad + c + 1, v.y * norm);
    atomicAdd(ad + c + 2, v.z * norm);
    atomicAdd(ad + c + 3, v.w * norm);
  }
}

// ---- fused self-loop + bias + ReLU, in place into agg ----------------------
__global__ void k_bias_relu_self(float* __restrict__ agg, const float* __restrict__ h,
                                 const float* __restrict__ dinv,
                                 const float* __restrict__ b, int N, int C) {
  long long idx = (long long)blockIdx.x * blockDim.x + threadIdx.x;
  if (idx >= (long long)N * C) return;
  int i = (int)(idx / C);
  int c = (int)(idx - (long long)i * C);
  float di = dinv[i];
  float v = agg[idx] + h[idx] * di * di + b[c];
  agg[idx] = v > 0.0f ? v : 0.0f;
}

// ---- final 8->1 projection -------------------------------------------------
__global__ void k_final(const float* __restrict__ h3, const float* __restrict__ Wf,
                        const float* __restrict__ bf, float* __restrict__ out, int N) {
  int i = blockIdx.x * blockDim.x + threadIdx.x;
  if (i >= N) return;
  float s = bf[0];
#pragma unroll
  for (int c = 0; c < 8; ++c) s += h3[i * 8 + c] * Wf[c];
  out[i] = s;
}

// ---------------------------------------------------------------------------
extern "C" void kernel_launch(void* const* d_in, const int* in_sizes, int n_in,
                              void* d_out, int out_size, void* d_ws, size_t ws_size,
                              hipStream_t stream) {
  const float*     x  = (const float*)d_in[0];
  const long long* ei = (const long long*)d_in[1];
  const float* W1 = (const float*)d_in[2]; const float* b1 = (const float*)d_in[3];
  const float* W2 = (const float*)d_in[4]; const float* b2 = (const float*)d_in[5];
  const float* W3 = (const float*)d_in[6]; const float* b3 = (const float*)d_in[7];
  const float* Wf = (const float*)d_in[8]; const float* bf = (const float*)d_in[9];

  const int       N = in_sizes[0] / 3;      // 100000
  const long long E = in_sizes[1] / 2;      // 1600000
  const long long* src = ei;
  const long long* dst = ei + E;

  // workspace: dinv[N] | h[N*32] | agg[N*32] | xp[N*4] | W1p | W2p | W3p
  float* fws  = (float*)d_ws;
  float* dinv = fws;
  float* h    = dinv + N;
  float* agg  = h + (size_t)N * 32;
  float* xp   = agg + (size_t)N * 32;
  float* W1p  = xp + (size_t)N * 4;   // NT=2, KQ=1 -> 128 floats
  float* W2p  = W1p + 128;            // NT=1, KQ=8 -> 512 floats
  float* W3p  = W2p + 512;            // NT=1, KQ=4 -> 256 floats

  const int T = 256;
  auto blocksN = [&](long long n) { return (unsigned)((n + T - 1) / T); };

  // --- degree / dinv: once per launch, shared by all 3 layers ---
  k_fill<<<blocksN(N), T, 0, stream>>>(dinv, 1.0f, N);          // self loop
  k_deg<<<blocksN(E), T, 0, stream>>>(dst, dinv, E);
  k_rsqrt<<<blocksN(N), T, 0, stream>>>(dinv, N);

  // --- operand packing (tiny, once per launch) ---
  k_packX<<<blocksN((long long)N * 4), T, 0, stream>>>(x, xp, N);
  k_packW<<<1, 128, 0, stream>>>(W1, W1p,  3, 32, 1, 2);
  k_packW<<<2, 256, 0, stream>>>(W2, W2p, 32, 16, 8, 1);
  k_packW<<<1, 256, 0, stream>>>(W3, W3p, 16,  8, 4, 1);

  struct Layer { const float* Wp; const float* b; int K; int KQ; int C; int NT; };
  Layer L[3] = { {W1p, b1,  4, 1, 32, 2},
                 {W2p, b2, 32, 8, 16, 1},
                 {W3p, b3, 16, 4,  8, 1} };

  const float* in = xp;
  for (int l = 0; l < 3; ++l) {
    int K = L[l].K, KQ = L[l].KQ, C = L[l].C, NT = L[l].NT;
    int mtiles = (N + 15) / 16;
    long long gthread = (long long)mtiles * NT * 32;

    // GEMM first (reads `in`, which is `agg` for layers 2/3), THEN zero agg.
    k_gemm_wmma<<<blocksN(gthread), T, 0, stream>>>(in, L[l].Wp, h, N, K, C,
                                                    NT, mtiles, KQ);
    k_fill<<<blocksN((long long)N * C), T, 0, stream>>>(agg, 0.0f, N * C);
    k_scatter<<<blocksN(E), T, 0, stream>>>(src, dst, dinv, h, agg, E, C);
    k_bias_relu_self<<<blocksN((long long)N * C), T, 0, stream>>>(agg, h, dinv,
                                                                  L[l].b, N, C);
    in = agg;
  }

  k_final<<<blocksN(N), T, 0, stream>>>(agg, Wf, bf, (float*)d_out, N);
}